// CostVolumeManager_28432683500058
// MI455X (gfx1250) — compile-verified
//
#include <hip/hip_runtime.h>
#include <math.h>

// Problem constants (B=1)
#define S_VIEWS 8
#define C_CH    16
#define H_IMG   64
#define W_IMG   96
#define N_PIX   (H_IMG * W_IMG)
#define D_BINS  64
#define EPS_Z   1e-8f

typedef __attribute__((ext_vector_type(2))) float v2f;
typedef __attribute__((ext_vector_type(8))) float v8f;

// ---------------------------------------------------------------------------
// Transpose features to channel-contiguous layout:
//   srcT[s][y][x][c] = src[s][c][y][x]   (S*N*16 floats)
//   curT[y][x][c]    = cur[c][y][x]      (N*16 floats)
// ---------------------------------------------------------------------------
__global__ __launch_bounds__(256) void cv_transpose_kernel(
    const float* __restrict__ cur, const float* __restrict__ src,
    float* __restrict__ curT, float* __restrict__ srcT)
{
    int i = blockIdx.x * blockDim.x + threadIdx.x;
    const int total_src = S_VIEWS * C_CH * N_PIX;
    if (i < total_src) {
        int s   = i / (C_CH * N_PIX);
        int r   = i % (C_CH * N_PIX);
        int c   = r / N_PIX;
        int lin = r % N_PIX;
        srcT[(s * N_PIX + lin) * C_CH + c] = src[i];
    } else {
        int j = i - total_src;
        if (j < C_CH * N_PIX) {
            int c   = j / N_PIX;
            int lin = j % N_PIX;
            curT[lin * C_CH + c] = cur[j];
        }
    }
}

// ---------------------------------------------------------------------------
// Main cost-volume kernel. One wave32 handles one (depth bin, 16-pixel tile).
// A = cur feats (16 pix x 16 ch), B = weighted gathered src feats
// (16 ch x 16 pix cols). cost = diag( sum_{s,tap} A x (w .* B_tap) ),
// accumulated with chained V_WMMA_F32_16X16X4_F32 over 4 K-chunks.
// XP: true -> channel-contiguous transposed layouts, false -> raw layouts.
// ---------------------------------------------------------------------------
template<bool XP>
__global__ __launch_bounds__(256) void cost_volume_wmma_kernel(
    const float* __restrict__ cur,     // XP ? [N][16]        : [C][H][W]
    const float* __restrict__ src,     // XP ? [S][H][W][16]  : [S][C][H][W]
    const float* __restrict__ exts,    // [S][4][4]
    const float* __restrict__ Ks,      // [S][4][4]
    const float* __restrict__ invK,    // [4][4] (uses 3x3 block)
    const float* __restrict__ mind,    // [1]
    const float* __restrict__ maxd,    // [1]
    float* __restrict__ out)           // [2][D][H][W]
{
    const int lane = threadIdx.x & 31;
    const int wave = blockIdx.x * (blockDim.x >> 5) + (threadIdx.x >> 5);

    // tile decomposition: 6 x-tiles, H rows, D bins
    const int xt    = wave % (W_IMG / 16);
    const int y     = (wave / (W_IMG / 16)) % H_IMG;
    const int d     = wave / ((W_IMG / 16) * H_IMG);
    const int xbase = xt * 16;

    const int p      = lane & 15;        // pixel within tile (column / row id)
    const int half   = (lane >> 4) & 1;  // lane half -> K-pair offset
    const int c0base = half * 2;
    const int x      = xbase + p;

    // log-spaced depth for this bin
    const float mn    = mind[0];
    const float mx    = maxd[0];
    const float depth = expf(logf(mn) + logf(mx / mn) *
                             ((float)d / (float)(D_BINS - 1)));

    // camera ray and 3D point for this pixel
    const float px = (float)x + 0.5f;
    const float py = (float)y + 0.5f;
    const float rx = invK[0] * px + invK[1] * py + invK[2];
    const float ry = invK[4] * px + invK[5] * py + invK[6];
    const float rz = invK[8] * px + invK[9] * py + invK[10];
    const float Xc = depth * rx, Yc = depth * ry, Zc = depth * rz;

    // A fragments: cur features, lane half picks channel pair {c0, c0+1}
    v2f a[4];
    {
        const int lin = y * W_IMG + x;
#pragma unroll
        for (int kk = 0; kk < 4; ++kk) {
            const int c0 = kk * 4 + c0base;
            if (XP) {
                const float* ap = cur + lin * C_CH + c0;
                a[kk].x = ap[0];
                a[kk].y = ap[1];
            } else {
                a[kk].x = cur[c0 * N_PIX + lin];
                a[kk].y = cur[(c0 + 1) * N_PIX + lin];
            }
        }
    }

    v8f acc = {0.f, 0.f, 0.f, 0.f, 0.f, 0.f, 0.f, 0.f};

    for (int s = 0; s < S_VIEWS; ++s) {
        // P = (K_s @ E_s)[:3,:4], uniform across the wave (scalarized loads)
        const float* Km = Ks + s * 16;
        const float* E  = exts + s * 16;
        float P[12];
#pragma unroll
        for (int i = 0; i < 3; ++i)
#pragma unroll
            for (int j = 0; j < 4; ++j)
                P[i * 4 + j] = Km[i * 4 + 0] * E[0 * 4 + j]
                             + Km[i * 4 + 1] * E[1 * 4 + j]
                             + Km[i * 4 + 2] * E[2 * 4 + j]
                             + Km[i * 4 + 3] * E[3 * 4 + j];

        const float uw = P[0] * Xc + P[1] * Yc + P[2]  * Zc + P[3];
        const float vw = P[4] * Xc + P[5] * Yc + P[6]  * Zc + P[7];
        const float zw = P[8] * Xc + P[9] * Yc + P[10] * Zc + P[11];

        const float inv = 1.0f / (zw + EPS_Z);
        const float gx  = uw * inv - 0.5f;
        const float gy  = vw * inv - 0.5f;
        const float x0f = floorf(gx);
        const float y0f = floorf(gy);
        const float wx  = gx - x0f;
        const float wy  = gy - y0f;
        const bool  zok = (zw > 0.0f);

#pragma unroll
        for (int t = 0; t < 4; ++t) {
            const float xf = x0f + (float)(t & 1);
            const float yf = y0f + (float)(t >> 1);
            const bool valid = zok && (xf >= 0.0f) && (xf < (float)W_IMG)
                                   && (yf >= 0.0f) && (yf < (float)H_IMG);
            float wt = ((t & 1)  ? wx : 1.0f - wx) *
                       ((t >> 1) ? wy : 1.0f - wy);
            wt = valid ? wt : 0.0f;  // select (NaN-safe, unlike multiply)

            // clamped integer tap coords (NaN-safe via fminf/fmaxf)
            const int xi = (int)fminf(fmaxf(xf, 0.0f), (float)(W_IMG - 1));
            const int yi = (int)fminf(fmaxf(yf, 0.0f), (float)(H_IMG - 1));

            // B fragments: gathered src channel pairs, pre-scaled by weight
            v2f b[4];
            if (XP) {
                const float* bp = src + (((s * H_IMG + yi) * W_IMG + xi) * C_CH)
                                      + c0base;
#pragma unroll
                for (int kk = 0; kk < 4; ++kk) {
                    b[kk].x = bp[kk * 4 + 0] * wt;
                    b[kk].y = bp[kk * 4 + 1] * wt;
                }
            } else {
                const int lin = yi * W_IMG + xi;
                const float* bp = src + (size_t)s * C_CH * N_PIX + lin;
#pragma unroll
                for (int kk = 0; kk < 4; ++kk) {
                    const int c0 = kk * 4 + c0base;
                    b[kk].x = bp[(size_t)c0 * N_PIX] * wt;
                    b[kk].y = bp[(size_t)(c0 + 1) * N_PIX] * wt;
                }
            }

            // D = A x B + C over the 4 K-chunks (fp32 WMMA, full precision)
#pragma unroll
            for (int kk = 0; kk < 4; ++kk) {
                acc = __builtin_amdgcn_wmma_f32_16x16x4_f32(
                    false, a[kk], false, b[kk], (short)0, acc, false, false);
            }
        }
    }

    // Extract diagonal: pixel q sits at (VGPR q&7, lane q if q<8 else q+16)
    const bool isdiag = (lane < 8) || (lane >= 24);
    const int  pd     = (lane < 8) ? lane : (lane - 16);
    const int  q      = pd & 7;
    float val = acc[0];
    val = (q == 1) ? acc[1] : val;
    val = (q == 2) ? acc[2] : val;
    val = (q == 3) ? acc[3] : val;
    val = (q == 4) ? acc[4] : val;
    val = (q == 5) ? acc[5] : val;
    val = (q == 6) ? acc[6] : val;
    val = (q == 7) ? acc[7] : val;

    const int outbase = (d * H_IMG + y) * W_IMG + xbase;
    if (isdiag)
        out[outbase + pd] = val;
    // second output: broadcast depth planes
    if (lane < 16)
        out[D_BINS * N_PIX + outbase + lane] = depth;
}

// ---------------------------------------------------------------------------
extern "C" void kernel_launch(void* const* d_in, const int* in_sizes, int n_in,
                              void* d_out, int out_size, void* d_ws, size_t ws_size,
                              hipStream_t stream)
{
    (void)in_sizes; (void)n_in; (void)out_size;
    const float* cur  = (const float*)d_in[0];  // cur_feats   [1,16,64,96]
    const float* src  = (const float*)d_in[1];  // src_feats   [1,8,16,64,96]
    const float* exts = (const float*)d_in[2];  // src_extrinsics [1,8,4,4]
    const float* Ks   = (const float*)d_in[3];  // src_Ks         [1,8,4,4]
    const float* invK = (const float*)d_in[4];  // cur_invK       [1,4,4]
    const float* mind = (const float*)d_in[5];  // min_depth (1 elem)
    const float* maxd = (const float*)d_in[6];  // max_depth (1 elem)
    float* out = (float*)d_out;

    // main kernel geometry: D * H * (W/16) = 24576 waves, 8 waves/block
    const int nWaves  = D_BINS * H_IMG * (W_IMG / 16);
    const int nBlocks = nWaves / 8;

    const size_t needT = (size_t)(S_VIEWS * N_PIX * C_CH + N_PIX * C_CH)
                         * sizeof(float);
    if (ws_size >= needT) {
        float* srcT = (float*)d_ws;
        float* curT = srcT + (size_t)S_VIEWS * N_PIX * C_CH;
        const int totalT = S_VIEWS * C_CH * N_PIX + C_CH * N_PIX;
        cv_transpose_kernel<<<(totalT + 255) / 256, 256, 0, stream>>>(
            cur, src, curT, srcT);
        cost_volume_wmma_kernel<true><<<nBlocks, 256, 0, stream>>>(
            curT, srcT, exts, Ks, invK, mind, maxd, out);
    } else {
        cost_volume_wmma_kernel<false><<<nBlocks, 256, 0, stream>>>(
            cur, src, exts, Ks, invK, mind, maxd, out);
    }
}